// EarthMoverDistance_38843684225042
// MI455X (gfx1250) — compile-verified
//
#include <hip/hip_runtime.h>
#include <math.h>

// Earth Mover Distance via entropic Sinkhorn, fused tile kernels for gfx1250.
// B=8, N=2048, eps=0.005, 50 iterations. Cost tiles are recomputed on the fly
// with V_WMMA_F32_16X16X4_F32 (K=3 dot products of 3-D points), with the full
// scaled LSE argument folded into the WMMA accumulator.

#define BATCH 8
#define NPTS 2048
#define BN (BATCH * NPTS)
#define EPS 0.005f
#define ITERS 50
#define LOG2E 1.4426950408889634f
#define LN2 0.6931471805599453f
#define LN_N 7.6246189861593985f   // ln(2048)

typedef __attribute__((ext_vector_type(2))) float v2f;
typedef __attribute__((ext_vector_type(8))) float v8f;

// ---------------------------------------------------------------------------
// Precompute squared norms of both point clouds; zero the g potential.
// ---------------------------------------------------------------------------
__global__ void init_kernel(const float* __restrict__ p1,
                            const float* __restrict__ p2,
                            float* __restrict__ sqX,
                            float* __restrict__ sqY,
                            float* __restrict__ g) {
  int i = blockIdx.x * blockDim.x + threadIdx.x;
  if (i < BN) {
    float a0 = p1[3 * i], a1 = p1[3 * i + 1], a2 = p1[3 * i + 2];
    sqX[i] = a0 * a0 + a1 * a1 + a2 * a2;
    float b0 = p2[3 * i], b1 = p2[3 * i + 1], b2 = p2[3 * i + 2];
    sqY[i] = b0 * b0 + b1 * b1 + b2 * b2;
    g[i] = 0.0f;
  }
}

// ---------------------------------------------------------------------------
// One Sinkhorn half-step:
//   fout_i = -eps*ln(N) - eps * LSE_j((gin_j - C_ij)/eps)
// Rows come from rowPts, reduction runs over colPts. Symmetric under swap.
// Each wave32 owns 16 rows; WMMA output d[k] is directly
//   t = (gin_j - C_ij) * log2(e)/eps   (base-2 exponent, feeds v_exp_f32).
// ---------------------------------------------------------------------------
__launch_bounds__(128)
__global__ void sinkhorn_half(const float* __restrict__ rowPts,
                              const float* __restrict__ colPts,
                              const float* __restrict__ rowSq,
                              const float* __restrict__ colSq,
                              const float* __restrict__ gin,
                              float* __restrict__ fout) {
  __shared__ float sYx[NPTS], sYy[NPTS], sYz[NPTS], sCT[NPTS];
  const float ieps2 = LOG2E / EPS;          // log2(e)/eps
  const int b = blockIdx.x >> 5;            // 32 row-blocks per batch
  const int rblk = blockIdx.x & 31;
  const float* rp = rowPts + (size_t)b * NPTS * 3;
  const float* cp = colPts + (size_t)b * NPTS * 3;
  const float* rs = rowSq + (size_t)b * NPTS;
  const float* cs = colSq + (size_t)b * NPTS;
  const float* gb = gin + (size_t)b * NPTS;
  float* fb = fout + (size_t)b * NPTS;

  // Stage column points + column term (g_j - |y_j|^2)*ieps2 into LDS.
  for (int j = threadIdx.x; j < NPTS; j += blockDim.x) {
    sYx[j] = cp[3 * j];
    sYy[j] = cp[3 * j + 1];
    sYz[j] = cp[3 * j + 2];
    sCT[j] = (gb[j] - cs[j]) * ieps2;
  }
  __syncthreads();

  const int lane = threadIdx.x & 31;
  const int wave = threadIdx.x >> 5;
  const int r0 = rblk * 64 + wave * 16;     // this wave's 16 rows
  const bool hi = lane >= 16;
  const int lh = lane & 15;

  // A fragment (16x4 f32): lanes 0-15 hold {x,y} (K=0,1) of row M=lane,
  // lanes 16-31 hold {z,0} (K=2,3) of row M=lane-16.
  v2f afrag;
  {
    const int mrow = r0 + lh;
    afrag.x = hi ? rp[3 * mrow + 2] : rp[3 * mrow];
    afrag.y = hi ? 0.0f : rp[3 * mrow + 1];
  }
  // Row term per accumulator slot: |x_i|^2 * ieps2 for rows k / k+8.
  float rt[8];
#pragma unroll
  for (int k = 0; k < 8; ++k) rt[k] = rs[r0 + k + (hi ? 8 : 0)] * ieps2;

  // Online (branchless) base-2 log-sum-exp state per lane slot.
  float m[8], s[8];
#pragma unroll
  for (int k = 0; k < 8; ++k) { m[k] = -1e30f; s[k] = 0.0f; }

  const float s2 = 2.0f * ieps2;
  for (int jt = 0; jt < NPTS / 16; ++jt) {
    const int j = jt * 16 + lh;
    // B fragment (4x16 f32), scaled by 2*ieps2 so WMMA emits the exponent.
    v2f bfrag;
    bfrag.x = (hi ? sYz[j] : sYx[j]) * s2;
    bfrag.y = (hi ? 0.0f : sYy[j]) * s2;
    const float ct = sCT[j];
    v8f acc;
#pragma unroll
    for (int k = 0; k < 8; ++k) acc[k] = ct - rt[k];
    // d[k] = (g_j - C_ij) * log2(e)/eps for (i = r0+k(+8), j = jt*16+lh)
    v8f d = __builtin_amdgcn_wmma_f32_16x16x4_f32(
        false, afrag, false, bfrag, (short)0, acc, false, false);
#pragma unroll
    for (int k = 0; k < 8; ++k) {
      const float t = d[k];
      const float mn = fmaxf(m[k], t);
      s[k] = s[k] * __builtin_amdgcn_exp2f(m[k] - mn) +
             __builtin_amdgcn_exp2f(t - mn);
      m[k] = mn;
    }
  }

  // LSE-combine across the 16 lanes of each half (columns of the same rows).
#pragma unroll
  for (int k = 0; k < 8; ++k) {
#pragma unroll
    for (int off = 1; off < 16; off <<= 1) {
      const float m2 = __shfl_xor(m[k], off, 32);
      const float s2v = __shfl_xor(s[k], off, 32);
      const float mn = fmaxf(m[k], m2);
      s[k] = s[k] * __builtin_amdgcn_exp2f(m[k] - mn) +
             s2v * __builtin_amdgcn_exp2f(m2 - mn);
      m[k] = mn;
    }
  }

  if (lh == 0) {
    const float negEpsLn2 = -EPS * LN2;
    const float cA = -EPS * LN_N;   // eps * log(1/N)
    const int rbase = r0 + (hi ? 8 : 0);
#pragma unroll
    for (int k = 0; k < 8; ++k) {
      // f = eps*log(1/N) - eps*ln2*(m + log2(s));  v_log_f32 is base-2.
      fb[rbase + k] = cA + negEpsLn2 * (m[k] + __builtin_amdgcn_logf(s[k]));
    }
  }
}

// ---------------------------------------------------------------------------
// Final per-point distance:
//   dist_i = N * sum_j 2^((f_i + g_j - C_ij)*log2e/eps) * C_ij
// Same WMMA tiling; C is recovered from the exponent algebraically:
//   C_ij = f_i + g_j - t * eps * ln2.
// Writes sqrt(dist_i) for the mean reduction.
// ---------------------------------------------------------------------------
__launch_bounds__(128)
__global__ void dist_kernel(const float* __restrict__ p1,
                            const float* __restrict__ p2,
                            const float* __restrict__ sqX,
                            const float* __restrict__ sqY,
                            const float* __restrict__ f,
                            const float* __restrict__ g,
                            float* __restrict__ distOut) {
  __shared__ float sYx[NPTS], sYy[NPTS], sYz[NPTS], sCT[NPTS], sG[NPTS];
  const float ieps2 = LOG2E / EPS;
  const int b = blockIdx.x >> 5;
  const int rblk = blockIdx.x & 31;
  const float* rp = p1 + (size_t)b * NPTS * 3;
  const float* cp = p2 + (size_t)b * NPTS * 3;
  const float* rs = sqX + (size_t)b * NPTS;
  const float* cs = sqY + (size_t)b * NPTS;
  const float* fbp = f + (size_t)b * NPTS;
  const float* gb = g + (size_t)b * NPTS;
  float* db = distOut + (size_t)b * NPTS;

  for (int j = threadIdx.x; j < NPTS; j += blockDim.x) {
    sYx[j] = cp[3 * j];
    sYy[j] = cp[3 * j + 1];
    sYz[j] = cp[3 * j + 2];
    const float gj = gb[j];
    sG[j] = gj;
    sCT[j] = (gj - cs[j]) * ieps2;
  }
  __syncthreads();

  const int lane = threadIdx.x & 31;
  const int wave = threadIdx.x >> 5;
  const int r0 = rblk * 64 + wave * 16;
  const bool hi = lane >= 16;
  const int lh = lane & 15;

  v2f afrag;
  {
    const int mrow = r0 + lh;
    afrag.x = hi ? rp[3 * mrow + 2] : rp[3 * mrow];
    afrag.y = hi ? 0.0f : rp[3 * mrow + 1];
  }
  float rt[8], fk[8];
#pragma unroll
  for (int k = 0; k < 8; ++k) {
    const int r = r0 + k + (hi ? 8 : 0);
    const float fv = fbp[r];
    fk[k] = fv;
    rt[k] = (rs[r] - fv) * ieps2;    // accumulator gets (f_i - |x_i|^2)*ieps2
  }

  float sum[8];
#pragma unroll
  for (int k = 0; k < 8; ++k) sum[k] = 0.0f;

  const float s2 = 2.0f * ieps2;
  const float epsLn2 = EPS * LN2;
  for (int jt = 0; jt < NPTS / 16; ++jt) {
    const int j = jt * 16 + lh;
    v2f bfrag;
    bfrag.x = (hi ? sYz[j] : sYx[j]) * s2;
    bfrag.y = (hi ? 0.0f : sYy[j]) * s2;
    const float ct = sCT[j];
    const float gj = sG[j];
    v8f acc;
#pragma unroll
    for (int k = 0; k < 8; ++k) acc[k] = ct - rt[k];
    v8f d = __builtin_amdgcn_wmma_f32_16x16x4_f32(
        false, afrag, false, bfrag, (short)0, acc, false, false);
#pragma unroll
    for (int k = 0; k < 8; ++k) {
      const float t = d[k];                 // (f+g-C)*log2e/eps  (= log2 P*..)
      const float C = fk[k] + gj - t * epsLn2;
      sum[k] += __builtin_amdgcn_exp2f(t) * C;
    }
  }

#pragma unroll
  for (int k = 0; k < 8; ++k) {
#pragma unroll
    for (int off = 1; off < 16; off <<= 1)
      sum[k] += __shfl_xor(sum[k], off, 32);
  }

  if (lh == 0) {
    const int rbase = r0 + (hi ? 8 : 0);
#pragma unroll
    for (int k = 0; k < 8; ++k) {
      const float dist = (float)NPTS * sum[k];
      db[rbase + k] = sqrtf(fmaxf(dist, 0.0f));
    }
  }
}

// ---------------------------------------------------------------------------
// Deterministic mean over B*N sqrt-distances (single block tree reduction).
// ---------------------------------------------------------------------------
__global__ void reduce_kernel(const float* __restrict__ dist,
                              float* __restrict__ out) {
  __shared__ float red[256];
  float acc = 0.0f;
  for (int i = threadIdx.x; i < BN; i += 256) acc += dist[i];
  red[threadIdx.x] = acc;
  __syncthreads();
  for (int st = 128; st > 0; st >>= 1) {
    if (threadIdx.x < (unsigned)st) red[threadIdx.x] += red[threadIdx.x + st];
    __syncthreads();
  }
  if (threadIdx.x == 0) out[0] = red[0] / (float)BN;
}

// ---------------------------------------------------------------------------
extern "C" void kernel_launch(void* const* d_in, const int* in_sizes, int n_in,
                              void* d_out, int out_size, void* d_ws,
                              size_t ws_size, hipStream_t stream) {
  (void)in_sizes; (void)n_in; (void)out_size; (void)ws_size;
  const float* p1 = (const float*)d_in[0];   // [B,N,3] fp32
  const float* p2 = (const float*)d_in[1];   // [B,N,3] fp32
  float* ws = (float*)d_ws;                  // 5*B*N floats = 320 KB
  float* g = ws;
  float* f = ws + BN;
  float* sqX = ws + 2 * BN;
  float* sqY = ws + 3 * BN;
  float* dst = ws + 4 * BN;

  const dim3 sgrid(BATCH * (NPTS / 64));     // 256 blocks
  const dim3 sblk(128);                      // 4 wave32 per block

  init_kernel<<<BN / 256, 256, 0, stream>>>(p1, p2, sqX, sqY, g);
  for (int it = 0; it < ITERS; ++it) {
    sinkhorn_half<<<sgrid, sblk, 0, stream>>>(p1, p2, sqX, sqY, g, f);
    sinkhorn_half<<<sgrid, sblk, 0, stream>>>(p2, p1, sqY, sqX, f, g);
  }
  sinkhorn_half<<<sgrid, sblk, 0, stream>>>(p1, p2, sqX, sqY, g, f);
  dist_kernel<<<sgrid, sblk, 0, stream>>>(p1, p2, sqX, sqY, f, g, dst);
  reduce_kernel<<<1, 256, 0, stream>>>(dst, (float*)d_out);
}